// DynamicBlockSVDLinear_80272938762887
// MI455X (gfx1250) — compile-verified
//
#include <hip/hip_runtime.h>

typedef __attribute__((ext_vector_type(2))) float v2f;
typedef __attribute__((ext_vector_type(8))) float v8f;

#define TOKENS 2048
#define DIMD   512
#define HIDW   256
#define DKN    100
#define NCLS   10000
#define KBLK   8
#define RNK    64
#define CPER   1250

enum { MODE_NONE = 0, MODE_BIAS_RELU = 1, MODE_BIAS = 2, MODE_SEL_BIAS = 3 };

// Generic fp32 WMMA GEMM: C[M,N] = epilogue(A[M,K](lda) @ B[N,K]^T(ldb))
// Block tile 128x64, 8 waves arranged 4(M) x 2(N), wave tile 32x32.
// blockIdx.z indexes independent sub-GEMMs (block-SVD): A += z*aZ, B += z*bZ,
// output/bias/sel columns offset by z*colZ.
template <int MODE>
__global__ __launch_bounds__(256) void gemm_wmma_f32(
    const float* __restrict__ A, const float* __restrict__ B,
    float* __restrict__ C, const float* __restrict__ bias,
    const unsigned char* __restrict__ sel, int M, int N, int K, int lda,
    int ldb, int ldc, long aZ, long bZ, int colZ) {
  __shared__ float Al[128 * 20];  // 16 K-cols padded to 20 (bank-conflict break)
  __shared__ float Bl[64 * 20];

  const int tid = threadIdx.x;
  const int z = blockIdx.z;
  A += (long)z * aZ;
  B += (long)z * bZ;
  const int colOff = z * colZ;

  const int mBase = blockIdx.y * 128;
  const int nBase = blockIdx.x * 64;
  const int wave = tid >> 5;
  const int lane = tid & 31;
  const int lm = lane & 15;
  const int hlf = lane >> 4;  // lane half: selects K=2h / M=v+8h per ISA layout
  const int waveM = wave & 3;
  const int waveN = wave >> 2;

  v8f acc[2][2];
#pragma unroll
  for (int mi = 0; mi < 2; ++mi)
#pragma unroll
    for (int ni = 0; ni < 2; ++ni) acc[mi][ni] = (v8f)0.0f;

  for (int k0 = 0; k0 < K; k0 += 16) {
    // ---- stage A tile 128x16 (two float4 per thread, fully coalesced)
    {
      int idx = tid;
#pragma unroll
      for (int i = 0; i < 2; ++i, idx += 256) {
        int row = idx >> 2;
        int c4 = (idx & 3) << 2;
        float4 v = *(const float4*)(A + (long)(mBase + row) * lda + (k0 + c4));
        *(float4*)(&Al[row * 20 + c4]) = v;
      }
    }
    // ---- stage B tile 64x16 (one float4 per thread), guard N edge
    {
      int row = tid >> 2;
      int c4 = (tid & 3) << 2;
      float4 v = make_float4(0.f, 0.f, 0.f, 0.f);
      if (nBase + row < N)
        v = *(const float4*)(B + (long)(nBase + row) * ldb + (k0 + c4));
      *(float4*)(&Bl[row * 20 + c4]) = v;
    }
    // prefetch next K tiles into cache hierarchy (global_prefetch_b8)
    if (k0 + 16 < K) {
      __builtin_prefetch(A + (long)(mBase + (tid >> 2)) * lda + k0 + 16, 0, 1);
      if (nBase + (tid >> 2) < N)
        __builtin_prefetch(B + (long)(nBase + (tid >> 2)) * ldb + k0 + 16, 0, 1);
    }
    __syncthreads();

// ---- 4 WMMA K-steps of 4, 2x2 tiles each
#pragma unroll
    for (int kk = 0; kk < 16; kk += 4) {
      v2f aF[2], bF[2];
#pragma unroll
      for (int mi = 0; mi < 2; ++mi)
        aF[mi] = *(const v2f*)(&Al[(waveM * 32 + mi * 16 + lm) * 20 + kk + 2 * hlf]);
#pragma unroll
      for (int ni = 0; ni < 2; ++ni)
        bF[ni] = *(const v2f*)(&Bl[(waveN * 32 + ni * 16 + lm) * 20 + kk + 2 * hlf]);
#pragma unroll
      for (int mi = 0; mi < 2; ++mi)
#pragma unroll
        for (int ni = 0; ni < 2; ++ni)
          acc[mi][ni] = __builtin_amdgcn_wmma_f32_16x16x4_f32(
              false, aF[mi], false, bF[ni], (short)0, acc[mi][ni], false, false);
    }
    __syncthreads();
  }

// ---- epilogue: C layout = VGPR v, lane-half h -> element C[v+8h][lm]
#pragma unroll
  for (int mi = 0; mi < 2; ++mi) {
#pragma unroll
    for (int ni = 0; ni < 2; ++ni) {
      int col = nBase + waveN * 32 + ni * 16 + lm;
      if (col < N) {
        int gcol = colOff + col;
        float bv = (MODE != MODE_NONE) ? bias[gcol] : 0.0f;
        bool s = (MODE == MODE_SEL_BIAS) ? (sel[gcol] != 0) : true;
        int rowT = mBase + waveM * 32 + mi * 16 + hlf * 8;
#pragma unroll
        for (int v = 0; v < 8; ++v) {
          float val = acc[mi][ni][v];
          if (MODE == MODE_BIAS_RELU)
            val = fmaxf(val + bv, 0.0f);
          else if (MODE == MODE_BIAS)
            val = val + bv;
          else if (MODE == MODE_SEL_BIAS)
            val = s ? (val + bv) : bv;
          C[(long)(rowT + v) * ldc + gcol] = val;
        }
      }
    }
  }
}

__global__ __launch_bounds__(256) void zero_bytes(unsigned char* p, int n) {
  int i = blockIdx.x * blockDim.x + threadIdx.x;
  if (i < n) p[i] = 0;
}

// Full descending argsort via stable rank counting:
// rank(r) = #{r' : s[r'] > s[r]} + #{r' < r : s[r'] == s[r]}  (matches lax.top_k)
// selected class = idx_base[rank] + r, clipped.
__global__ __launch_bounds__(128) void candidate_rank_select(
    const float* __restrict__ score, const int* __restrict__ idx_base,
    unsigned char* __restrict__ sel) {
  __shared__ float ls[128 * DKN];
  const int tid = threadIdx.x;
  const long base = (long)blockIdx.x * 128 * DKN;
  for (int i = tid; i < 128 * DKN; i += 128) ls[i] = score[base + i];
  __syncthreads();
  const float* s = &ls[tid * DKN];
#pragma unroll 1
  for (int r = 0; r < DKN; ++r) {
    float sr = s[r];
    int rank = 0;
#pragma unroll 4
    for (int r2 = 0; r2 < DKN; ++r2) {
      float s2 = s[r2];
      rank += ((s2 > sr) || ((s2 == sr) && (r2 < r))) ? 1 : 0;
    }
    int cls = idx_base[rank] + r;
    cls = cls < 0 ? 0 : (cls > NCLS - 1 ? NCLS - 1 : cls);
    sel[cls] = 1;  // benign race: all writers store 1
  }
}

extern "C" void kernel_launch(void* const* d_in, const int* in_sizes, int n_in,
                              void* d_out, int out_size, void* d_ws,
                              size_t ws_size, hipStream_t stream) {
  (void)in_sizes; (void)n_in; (void)out_size; (void)ws_size;
  const float* x = (const float*)d_in[0];        // [2048,512]
  const float* W1 = (const float*)d_in[1];       // [256,512]
  const float* b1 = (const float*)d_in[2];       // [256]
  const float* W2 = (const float*)d_in[3];       // [100,256]
  const float* b2 = (const float*)d_in[4];       // [100]
  const float* U = (const float*)d_in[5];        // [8,64,512] == [512,512]
  const float* V = (const float*)d_in[6];        // [8,1250,64]
  const float* bias = (const float*)d_in[7];     // [10000]
  const int* idx_base = (const int*)d_in[8];     // [100]
  float* out = (float*)d_out;                    // [2048,10000]

  float* h = (float*)d_ws;                       // 2048*256
  float* score = h + (size_t)TOKENS * HIDW;      // 2048*100
  float* ur = score + (size_t)TOKENS * DKN;      // 2048*512
  unsigned char* sel = (unsigned char*)(ur + (size_t)TOKENS * DIMD);  // 10000 B

  dim3 blk(256);
  // 1) h = relu(x @ W1^T + b1)
  gemm_wmma_f32<MODE_BIAS_RELU><<<dim3(HIDW / 64, TOKENS / 128, 1), blk, 0, stream>>>(
      x, W1, h, b1, nullptr, TOKENS, HIDW, DIMD, DIMD, DIMD, HIDW, 0, 0, 0);
  // 2) score = h @ W2^T + b2
  gemm_wmma_f32<MODE_BIAS><<<dim3((DKN + 63) / 64, TOKENS / 128, 1), blk, 0, stream>>>(
      h, W2, score, b2, nullptr, TOKENS, DKN, HIDW, HIDW, HIDW, DKN, 0, 0, 0);
  // 3) membership mask
  zero_bytes<<<dim3((NCLS + 255) / 256), blk, 0, stream>>>(sel, NCLS);
  candidate_rank_select<<<dim3(TOKENS / 128), dim3(128), 0, stream>>>(score, idx_base, sel);
  // 4) ur = x @ Uflat^T   (Uflat = U viewed as [512,512])
  gemm_wmma_f32<MODE_NONE><<<dim3((KBLK * RNK) / 64, TOKENS / 128, 1), blk, 0, stream>>>(
      x, U, ur, nullptr, nullptr, TOKENS, KBLK * RNK, DIMD, DIMD, DIMD, KBLK * RNK, 0, 0, 0);
  // 5) out[:, z*1250 + c] = sel ? ur_z @ V_z^T + bias : bias   (z = 0..7)
  gemm_wmma_f32<MODE_SEL_BIAS><<<dim3((CPER + 63) / 64, TOKENS / 128, KBLK), blk, 0, stream>>>(
      ur, V, out, bias, sel, TOKENS, CPER, RNK, KBLK * RNK, RNK, NCLS,
      (long)RNK, (long)CPER * RNK, CPER);
}